// safe_softmax_40991167873104
// MI455X (gfx1250) — compile-verified
//
#include <hip/hip_runtime.h>

// Safe softmax over a flat fp32 vector (N = 2^25), tuned for MI455X (gfx1250).
//
// Roofline: mandatory traffic = 1 HBM read of x (128MB) + 1 HBM write of y
// (128MB) = 256MB -> ~11us @ 23.3 TB/s. The 192MB L2 caches all of x after
// pass 1, so the pass-2 and pass-3 re-reads hit L2, not HBM. Compute is
// trivial (2 * 2^25 v_exp_f32), so the kernel lives entirely on the memory
// roofline. WMMA is deliberately unused: no matrix structure, no byte savings.
//
// Structure (5 launches, one stream):
//   1. per-block partial max  (tiled, unrolled b128 loads, RT hint -> L2)
//   2. final max              (1 block)
//   3. per-block partial sum of exp(x - M)   (L2 hits)
//   4. final sum -> 1/S       (1 block)
//   5. y = exp(x - M) / S     (non-temporal last-use load + NT store)

typedef __attribute__((ext_vector_type(4))) float v4f;

#define NEG_INF (-__builtin_inff())
#define NT_THREADS 256u
#define ITERS 8u
#define TILE (NT_THREADS * ITERS)   // 2048 float4 = 32KB per block-tile

__device__ __forceinline__ float waveRedMax(float v) {
#pragma unroll
  for (int off = 16; off > 0; off >>= 1)
    v = fmaxf(v, __shfl_xor(v, off, 32));
  return v;
}

__device__ __forceinline__ float waveRedSum(float v) {
#pragma unroll
  for (int off = 16; off > 0; off >>= 1)
    v += __shfl_xor(v, off, 32);
  return v;
}

// Block reductions: up to 1024 threads = 32 wave32 waves.
__device__ __forceinline__ float blockRedMax(float v) {
  __shared__ float smem[32];
  const int lane = threadIdx.x & 31;
  const int wid  = threadIdx.x >> 5;
  const int nw   = (blockDim.x + 31) >> 5;
  v = waveRedMax(v);
  if (lane == 0) smem[wid] = v;
  __syncthreads();
  if (wid == 0) {
    float w = (lane < nw) ? smem[lane] : NEG_INF;
    w = waveRedMax(w);
    if (lane == 0) smem[0] = w;
  }
  __syncthreads();
  return smem[0];
}

__device__ __forceinline__ float blockRedSum(float v) {
  __shared__ float smem[32];
  const int lane = threadIdx.x & 31;
  const int wid  = threadIdx.x >> 5;
  const int nw   = (blockDim.x + 31) >> 5;
  v = waveRedSum(v);
  if (lane == 0) smem[wid] = v;
  __syncthreads();
  if (wid == 0) {
    float w = (lane < nw) ? smem[lane] : 0.0f;
    w = waveRedSum(w);
    if (lane == 0) smem[0] = w;
  }
  __syncthreads();
  return smem[0];
}

// ---------------- Pass 1: per-block partial max ----------------
__global__ void k_partial_max(const float* __restrict__ x,
                              float* __restrict__ pmax,
                              unsigned n4, unsigned ntiles, long long n) {
  const v4f* __restrict__ x4 = (const v4f*)x;
  float m = NEG_INF;
  for (unsigned tile = blockIdx.x; tile < ntiles; tile += gridDim.x) {
    const unsigned base = tile * TILE + threadIdx.x;
    if ((tile + 1u) * TILE <= n4) {           // full tile: no per-load bounds
#pragma unroll
      for (unsigned j = 0; j < ITERS; ++j) {
        v4f v = x4[base + j * NT_THREADS];
        m = fmaxf(m, fmaxf(fmaxf(v.x, v.y), fmaxf(v.z, v.w)));
      }
    } else {                                   // edge tile: guarded
#pragma unroll
      for (unsigned j = 0; j < ITERS; ++j) {
        const unsigned idx = base + j * NT_THREADS;
        if (idx < n4) {
          v4f v = x4[idx];
          m = fmaxf(m, fmaxf(fmaxf(v.x, v.y), fmaxf(v.z, v.w)));
        }
      }
    }
  }
  if (blockIdx.x == 0 && threadIdx.x == 0) {   // scalar tail (N % 4)
    for (long long i = ((long long)n4) << 2; i < n; ++i) m = fmaxf(m, x[i]);
  }
  m = blockRedMax(m);
  if (threadIdx.x == 0) pmax[blockIdx.x] = m;
}

// ---------------- Pass 1b: reduce partial maxima -> M ----------------
__global__ void k_final_max(const float* __restrict__ pmax, int nb,
                            float* __restrict__ Mout) {
  float m = NEG_INF;
  for (int i = threadIdx.x; i < nb; i += blockDim.x) m = fmaxf(m, pmax[i]);
  m = blockRedMax(m);
  if (threadIdx.x == 0) *Mout = m;
}

// ---------------- Pass 2: per-block partial sum of exp(x - M) ----------------
__global__ void k_partial_sum(const float* __restrict__ x,
                              const float* __restrict__ Mp,
                              float* __restrict__ psum,
                              unsigned n4, unsigned ntiles, long long n) {
  const float M = *Mp;                          // uniform -> scalar load
  const v4f* __restrict__ x4 = (const v4f*)x;
  float s0 = 0.0f, s1 = 0.0f;                   // dual accumulators for ILP
  for (unsigned tile = blockIdx.x; tile < ntiles; tile += gridDim.x) {
    const unsigned base = tile * TILE + threadIdx.x;
    if ((tile + 1u) * TILE <= n4) {
#pragma unroll
      for (unsigned j = 0; j < ITERS; ++j) {
        v4f v = x4[base + j * NT_THREADS];
        s0 += __expf(v.x - M) + __expf(v.y - M);
        s1 += __expf(v.z - M) + __expf(v.w - M);
      }
    } else {
#pragma unroll
      for (unsigned j = 0; j < ITERS; ++j) {
        const unsigned idx = base + j * NT_THREADS;
        if (idx < n4) {
          v4f v = x4[idx];
          s0 += __expf(v.x - M) + __expf(v.y - M);
          s1 += __expf(v.z - M) + __expf(v.w - M);
        }
      }
    }
  }
  float s = s0 + s1;
  if (blockIdx.x == 0 && threadIdx.x == 0) {
    for (long long i = ((long long)n4) << 2; i < n; ++i) s += __expf(x[i] - M);
  }
  s = blockRedSum(s);
  if (threadIdx.x == 0) psum[blockIdx.x] = s;
}

// ---------------- Pass 2b: reduce partial sums -> 1/S ----------------
__global__ void k_final_sum(const float* __restrict__ psum, int nb,
                            float* __restrict__ invSout) {
  float s = 0.0f;
  for (int i = threadIdx.x; i < nb; i += blockDim.x) s += psum[i];
  s = blockRedSum(s);
  if (threadIdx.x == 0) *invSout = 1.0f / s;
}

// ---------------- Pass 3: y = exp(x - M) * (1/S) ----------------
// Non-temporal (last-use) loads of x and non-temporal stores of y: x is dead
// after this pass, and y is never re-read, so neither should occupy L2.
__global__ void k_normalize(const float* __restrict__ x,
                            const float* __restrict__ Mp,
                            const float* __restrict__ invSp,
                            float* __restrict__ y,
                            unsigned n4, unsigned ntiles, long long n) {
  const float M = *Mp;
  const float r = *invSp;
  const v4f* __restrict__ x4 = (const v4f*)x;
  v4f* __restrict__ y4 = (v4f*)y;
  for (unsigned tile = blockIdx.x; tile < ntiles; tile += gridDim.x) {
    const unsigned base = tile * TILE + threadIdx.x;
    if ((tile + 1u) * TILE <= n4) {
#pragma unroll
      for (unsigned j = 0; j < ITERS; ++j) {
        const unsigned idx = base + j * NT_THREADS;
        v4f v = __builtin_nontemporal_load(x4 + idx);
        v4f o;
        o.x = __expf(v.x - M) * r;
        o.y = __expf(v.y - M) * r;
        o.z = __expf(v.z - M) * r;
        o.w = __expf(v.w - M) * r;
        __builtin_nontemporal_store(o, y4 + idx);
      }
    } else {
#pragma unroll
      for (unsigned j = 0; j < ITERS; ++j) {
        const unsigned idx = base + j * NT_THREADS;
        if (idx < n4) {
          v4f v = __builtin_nontemporal_load(x4 + idx);
          v4f o;
          o.x = __expf(v.x - M) * r;
          o.y = __expf(v.y - M) * r;
          o.z = __expf(v.z - M) * r;
          o.w = __expf(v.w - M) * r;
          __builtin_nontemporal_store(o, y4 + idx);
        }
      }
    }
  }
  if (blockIdx.x == 0 && threadIdx.x == 0) {
    for (long long i = ((long long)n4) << 2; i < n; ++i)
      __builtin_nontemporal_store(__expf(x[i] - M) * r, y + i);
  }
}

extern "C" void kernel_launch(void* const* d_in, const int* in_sizes, int n_in,
                              void* d_out, int out_size, void* d_ws, size_t ws_size,
                              hipStream_t stream) {
  const float* x = (const float*)d_in[0];
  float* y = (float*)d_out;
  const long long n = (long long)in_sizes[0];
  const unsigned n4 = (unsigned)(n >> 2);                 // 8,388,608 float4
  const unsigned ntiles = (n4 + TILE - 1u) / TILE;        // 4096 tiles of 32KB

  // One block per tile if workspace allows (4096 blocks x 8 wave32 waves
  // saturates HBM); otherwise fewer blocks loop over tiles.
  unsigned nb = ntiles < 4096u ? (ntiles ? ntiles : 1u) : 4096u;
  while ((size_t)(2 + 2 * (size_t)nb) * sizeof(float) > ws_size && nb > 64u)
    nb >>= 1;

  float* ws   = (float*)d_ws;
  float* Mp   = ws + 0;        // global max
  float* iSp  = ws + 1;        // 1 / sum(exp(x - M))
  float* pmax = ws + 2;        // nb partial maxima
  float* psum = ws + 2 + nb;   // nb partial sums

  k_partial_max<<<nb, NT_THREADS, 0, stream>>>(x, pmax, n4, ntiles, n);
  k_final_max  <<<1, 1024, 0, stream>>>(pmax, (int)nb, Mp);
  k_partial_sum<<<nb, NT_THREADS, 0, stream>>>(x, Mp, psum, n4, ntiles, n);
  k_final_sum  <<<1, 1024, 0, stream>>>(psum, (int)nb, iSp);
  k_normalize  <<<nb, NT_THREADS, 0, stream>>>(x, Mp, iSp, y, n4, ntiles, n);
}